// PRALoss_43233140801681
// MI455X (gfx1250) — compile-verified
//
#include <hip/hip_runtime.h>

// PRA loss for D_est, D_gt of shape [32, 1, 1024, 1024] (f32).
// Bandwidth-bound (268 MB read once -> ~11.5us floor at 23.3 TB/s).
// Batch sums for the mask are computed with V_WMMA_F32_16X16X4_F32
// (full f32 precision); diff2/G2/H2 partials with VALU; deterministic
// two-stage tree reduction (no float atomics).

typedef __attribute__((ext_vector_type(2))) float v2f;
typedef __attribute__((ext_vector_type(8))) float v8f;

#define PRA_LAMBDA 0.1f
#define PRA_BATCH  32

__global__ __launch_bounds__(256) void pra_partial_kernel(
    const float* __restrict__ est, const float* __restrict__ gt,
    float* __restrict__ partial, int npix)
{
  const int lane        = threadIdx.x & 31;   // wave32
  const int wave        = threadIdx.x >> 5;
  const int wavesPerBlk = blockDim.x >> 5;
  const int gwave       = blockIdx.x * wavesPerBlk + wave;
  const int nwaves      = gridDim.x * wavesPerBlk;
  const int ntiles      = npix >> 4;          // 16 pixels per tile

  const int col  = lane & 15;                 // pixel within tile
  const int half = lane >> 4;                 // 0: K=0,1   1: K=2,3

  const v2f ones = {1.0f, 1.0f};              // A = ones(16x4), all lanes {1,1}

  float diff2 = 0.0f, g2 = 0.0f, h2 = 0.0f;

  // Loop bound is wave-uniform -> EXEC all-ones at every WMMA (required).
  for (int tile = gwave; tile < ntiles; tile += nwaves) {
    const size_t p = (size_t)tile * 16 + (size_t)col;
    v8f ce = {};                              // accumulates s_est per pixel
    v8f cg = {};                              // accumulates s_gt  per pixel
    float p2t = 0.0f;                         // this lane's sum of e^2 (its batches, its pixel)
#pragma unroll
    for (int i = 0; i < 8; ++i) {             // 8 * K=4 = 32 batches
      const int b0 = 4 * i + 2 * half;
      const float e0 = est[(size_t)b0       * (size_t)npix + p];
      const float e1 = est[(size_t)(b0 + 1) * (size_t)npix + p];
      const float g0 = gt [(size_t)b0       * (size_t)npix + p];
      const float g1 = gt [(size_t)(b0 + 1) * (size_t)npix + p];

      const float d0 = e0 - g0, d1 = e1 - g1;
      diff2 += d0 * d0 + d1 * d1;
      p2t   += e0 * e0 + e1 * e1;

      // B operand (4x16 f32): lanes 0-15 hold rows K=0 (v.x) / K=1 (v.y),
      // lanes 16-31 hold rows K=2 (v.x) / K=3 (v.y) -- matches b0 above.
      v2f be = {e0, e1};
      v2f bg = {g0, g1};
      ce = __builtin_amdgcn_wmma_f32_16x16x4_f32(false, ones, false, be,
                                                 (short)0, ce, false, false);
      cg = __builtin_amdgcn_wmma_f32_16x16x4_f32(false, ones, false, bg,
                                                 (short)0, cg, false, false);
    }
    g2 += p2t;

    // Per-pixel sum of est^2: combine the two half-wave partials.
    const float p2pix = p2t + __shfl_xor(p2t, 16, 32);

    // All D rows are identical column sums; element 0 in lane n is the
    // batch sum for pixel (tile*16 + (n&15)).
    const bool mask = ce[0] > cg[0];
    h2 += (mask && (half == 0)) ? p2pix : 0.0f;   // count each pixel once
  }

  // Wave32 tree reduce (deterministic).
  for (int off = 16; off >= 1; off >>= 1) {
    diff2 += __shfl_xor(diff2, off, 32);
    g2    += __shfl_xor(g2,    off, 32);
    h2    += __shfl_xor(h2,    off, 32);
  }

  __shared__ float sred[3][8];
  if (lane == 0) { sred[0][wave] = diff2; sred[1][wave] = g2; sred[2][wave] = h2; }
  __syncthreads();
  if (threadIdx.x == 0) {
    float a = 0.0f, b = 0.0f, c = 0.0f;
    for (int w = 0; w < wavesPerBlk; ++w) {
      a += sred[0][w]; b += sred[1][w]; c += sred[2][w];
    }
    partial[blockIdx.x * 3 + 0] = a;
    partial[blockIdx.x * 3 + 1] = b;
    partial[blockIdx.x * 3 + 2] = c;
  }
}

__global__ __launch_bounds__(256) void pra_final_kernel(
    const float* __restrict__ partial, int nb, float* __restrict__ out)
{
  __shared__ float sd[256], sg[256], sh[256];
  float a = 0.0f, b = 0.0f, c = 0.0f;
  for (int i = threadIdx.x; i < nb; i += blockDim.x) {
    a += partial[i * 3 + 0];
    b += partial[i * 3 + 1];
    c += partial[i * 3 + 2];
  }
  const int t = threadIdx.x;
  sd[t] = a; sg[t] = b; sh[t] = c;
  __syncthreads();
  for (int s = 128; s > 0; s >>= 1) {
    if (t < s) { sd[t] += sd[t + s]; sg[t] += sg[t + s]; sh[t] += sh[t + s]; }
    __syncthreads();
  }
  if (t == 0) {
    const float diff2 = sd[0], G2 = sg[0], H2 = sh[0];
    out[0] = diff2 / G2 + PRA_LAMBDA * (diff2 / H2);
  }
}

extern "C" void kernel_launch(void* const* d_in, const int* in_sizes, int n_in,
                              void* d_out, int out_size, void* d_ws, size_t ws_size,
                              hipStream_t stream) {
  (void)n_in; (void)out_size;
  const float* est = (const float*)d_in[0];
  const float* gt  = (const float*)d_in[1];
  float* out       = (float*)d_out;
  float* partial   = (float*)d_ws;

  const int npix = in_sizes[0] / PRA_BATCH;   // 1024*1024

  int nb = 2048;                              // 2048 blocks * 12 B = 24 KB of d_ws
  const size_t maxb = ws_size / (3 * sizeof(float));
  if ((size_t)nb > maxb) nb = (int)maxb;
  if (nb < 1) nb = 1;

  pra_partial_kernel<<<nb, 256, 0, stream>>>(est, gt, partial, npix);
  pra_final_kernel<<<1, 256, 0, stream>>>(partial, nb, out);
}